// RoIPooling_65532611002744
// MI455X (gfx1250) — compile-verified
//
#include <hip/hip_runtime.h>

// RoI crop_and_resize (bilinear, extrapolation=0), faithful to
// tf.image.crop_and_resize as ported in the JAX reference.
//
// feature_maps: [8,100,100,256] f32   rois: [2000,5] f32 (b, x1,y1,x2,y2 in px)
// out:          [2000,7,7,256]  f32
//
// Memory-bound gather (AI ~0.45 flop/byte: 9 flops per 20 B per channel): one 64-thread (2-wave32) block
// per output pixel, lane l handles channels [4l,4l+4) -> 4x global_load_b128
// (corners), bilinear blend in VALU, nontemporal b128 store. The 102.4 MB
// feature map fits in the 192 MB L2, so NT output stores (write-once 100 MB)
// preserve its residency -> ~203 MB HBM traffic ~= 9 us at 23.3 TB/s.
//
// 3D grid (ox, oy, roi): block coordinates come straight from workgroup IDs,
// eliminating the magic-number integer divides seen in the round-1 SALU mix.

typedef float v4f __attribute__((ext_vector_type(4)));

namespace {
constexpr int FM_H  = 100;
constexpr int FM_W  = 100;
constexpr int FM_C  = 256;
constexpr int OUT_H = 7;
constexpr int OUT_W = 7;
}

__global__ __launch_bounds__(64) void roi_crop_resize_kernel(
    const float* __restrict__ fm,
    const float* __restrict__ rois,
    float* __restrict__ out)
{
    const int ox = blockIdx.x;       // 0..6
    const int oy = blockIdx.y;       // 0..6
    const int n  = blockIdx.z;       // roi index

    // ROI record is block-uniform -> scalar s_load path.
    const float* __restrict__ r = rois + (size_t)n * 5;
    const int   b  = (int)r[0];
    const float x1 = r[1] * (1.0f / FM_W);
    const float y1 = r[2] * (1.0f / FM_H);
    const float x2 = r[3] * (1.0f / FM_W);
    const float y2 = r[4] * (1.0f / FM_H);

    // Sample coordinates (reciprocal-multiply instead of v_div_scale chain).
    constexpr float inv_oh = 1.0f / (float)(OUT_H - 1);
    constexpr float inv_ow = 1.0f / (float)(OUT_W - 1);
    const float in_y = y1 * (float)(FM_H - 1)
                     + (float)oy * ((y2 - y1) * (float)(FM_H - 1) * inv_oh);
    const float in_x = x1 * (float)(FM_W - 1)
                     + (float)ox * ((x2 - x1) * (float)(FM_W - 1) * inv_ow);

    const bool valid = (in_y >= 0.0f) && (in_y <= (float)(FM_H - 1)) &&
                       (in_x >= 0.0f) && (in_x <= (float)(FM_W - 1));

    const float fy = floorf(in_y);
    const float fx = floorf(in_x);
    const float ly = in_y - fy;
    const float lx = in_x - fx;

    const int ty = min(max((int)fy,          0), FM_H - 1);
    const int by = min(max((int)ceilf(in_y), 0), FM_H - 1);
    const int tx = min(max((int)fx,          0), FM_W - 1);
    const int bx = min(max((int)ceilf(in_x), 0), FM_W - 1);

    // Lane -> 4 consecutive channels; fully coalesced b128 per corner.
    const int    c    = (int)threadIdx.x * 4;
    const size_t rowT = ((size_t)b * FM_H + ty) * (size_t)FM_W;
    const size_t rowB = ((size_t)b * FM_H + by) * (size_t)FM_W;

    // Four independent 16B loads -> 4 outstanding LOADcnt before the wait.
    const v4f tl = *(const v4f*)(fm + (rowT + (size_t)tx) * FM_C + c);
    const v4f tr = *(const v4f*)(fm + (rowT + (size_t)bx) * FM_C + c);
    const v4f bl = *(const v4f*)(fm + (rowB + (size_t)tx) * FM_C + c);
    const v4f br = *(const v4f*)(fm + (rowB + (size_t)bx) * FM_C + c);

    const v4f top = tl + (tr - tl) * lx;
    const v4f bot = bl + (br - bl) * lx;
    v4f o         = top + (bot - top) * ly;
    if (!valid) o = v4f{0.0f, 0.0f, 0.0f, 0.0f};

    // Streaming (write-once) output: nontemporal store protects L2 residency
    // of the feature map.
    const size_t pix = ((size_t)n * OUT_H + oy) * OUT_W + ox;
    v4f* __restrict__ dst = (v4f*)(out + pix * FM_C + c);
    __builtin_nontemporal_store(o, dst);
}

extern "C" void kernel_launch(void* const* d_in, const int* in_sizes, int n_in,
                              void* d_out, int out_size, void* d_ws, size_t ws_size,
                              hipStream_t stream) {
    const float* fm   = (const float*)d_in[0];   // [8,100,100,256] f32
    const float* rois = (const float*)d_in[1];   // [N,5] f32
    float*       out  = (float*)d_out;           // [N,7,7,256] f32

    const int n_roi = in_sizes[1] / 5;

    dim3 grid(OUT_W, OUT_H, n_roi);              // one 2-wave block per pixel
    roi_crop_resize_kernel<<<grid, 64, 0, stream>>>(fm, rois, out);
}